// PR_23965917511945
// MI455X (gfx1250) — compile-verified
//
#include <hip/hip_runtime.h>

typedef float v2f __attribute__((ext_vector_type(2)));
typedef float v8f __attribute__((ext_vector_type(8)));

// Problem constants (reference: B=512, N=10, K=5, Q=5, H=256, Qtot=50)
#define HDIM 256
#define NCLS 10
#define KSH  5
#define QTOT 50
#define SQ   260   // LDS row stride in floats: (4n+k) mod 64 -> conflict-free 16-lane gathers

// LDS layout (float offsets)
#define OFF_QN  0                        // normalized query  [50][SQ]
#define OFF_SN  (OFF_QN + QTOT*SQ)       // normalized support[50][SQ]
#define OFF_M   (OFF_SN + QTOT*SQ)       // prototypes       [16][SQ] (rows 10..15 zero)
#define OFF_P   (OFF_M  + 16*SQ)         // rectified protos [16][SQ] (rows 10..15 zero)
#define OFF_DQM (OFF_P  + 16*SQ)         // q  . M^T  [64][16]
#define OFF_DSM (OFF_DQM + 64*16)        // s  . M^T  [64][16]
#define OFF_DQP (OFF_DSM + 64*16)        // q  . P^T  [64][16]
#define OFF_W   (OFF_DQP + 64*16)        // softmax weights [56][16] (slot-major)
#define OFF_QQ  (OFF_W  + 56*16)         // ||q||^2  [64]
#define OFF_MM  (OFF_QQ + 64)            // ||M||^2  [16]
#define OFF_PP  (OFF_MM + 16)            // ||P||^2  [16]
#define OFF_INV (OFF_PP + 16)            // 1/(K+count) [16]
#define OFF_CQ  (OFF_INV + 16)           // int Cq[64]
#define SMEM_FLOATS (OFF_CQ + 64)

__device__ __forceinline__ float wave_sum(float v) {
#pragma unroll
  for (int m = 16; m > 0; m >>= 1) v += __shfl_xor(v, m, 32);
  return v;
}

// One 16x16 output tile of A(64x256, padded rows) . B(16x256)^T using
// V_WMMA_F32_16X16X4_F32, K swept 256/4 = 64 steps.
// ISA layouts (CDNA5 ISA): A 16x4 f32: lane&15 = row, VGPR0/1 = K {0,1} (lanes 0-15)
// or K {2,3} (lanes 16-31). B 4x16 mirrors (lane&15 = column). D: 8 VGPRs,
// row = 8*(lane>>4) + j, col = lane&15.
__device__ __forceinline__ void gemm16(const float* __restrict__ A,
                                       const float* __restrict__ Bm,
                                       float* __restrict__ D,
                                       int tile, int lane) {
  const int m  = lane & 15;
  const int h2 = (lane >> 4) << 1;          // 0 or 2 : which K pair of the 4
  int rA = tile * 16 + m;
  rA = (rA > QTOT - 1) ? (QTOT - 1) : rA;   // clamp pad rows (results ignored)
  const float* ap = A  + rA * SQ + h2;
  const float* bp = Bm + m  * SQ + h2;
  v8f acc = {0.f, 0.f, 0.f, 0.f, 0.f, 0.f, 0.f, 0.f};
#pragma unroll 8
  for (int k0 = 0; k0 < HDIM; k0 += 4) {
    v2f av = *(const v2f*)(ap + k0);
    v2f bv = *(const v2f*)(bp + k0);
    acc = __builtin_amdgcn_wmma_f32_16x16x4_f32(false, av, false, bv,
                                                (short)0, acc, false, false);
  }
  const int orow = tile * 16 + ((lane >> 4) << 3);
#pragma unroll
  for (int j = 0; j < 8; ++j) D[(orow + j) * 16 + m] = acc[j];
}

__global__ __launch_bounds__(256) void PR_23965917511945_kernel(
    const float* __restrict__ support, const float* __restrict__ query,
    const float* __restrict__ gamma, const float* __restrict__ beta,
    float* __restrict__ out, int B) {
  extern __shared__ float sm[];
  int* Cq = (int*)&sm[OFF_CQ];
  const int tid  = threadIdx.x;
  const int lane = tid & 31;
  const int wave = tid >> 5;
  const int b    = blockIdx.x;

  if (tid < 16) { sm[OFF_MM + tid] = 0.f; sm[OFF_PP + tid] = 0.f; }

  // gamma/beta for this lane's 8 columns (h = lane*8 + j) — reused for every row
  const float4 g0 = *(const float4*)(gamma + lane * 8);
  const float4 g1 = *(const float4*)(gamma + lane * 8 + 4);
  const float4 e0 = *(const float4*)(beta  + lane * 8);
  const float4 e1 = *(const float4*)(beta  + lane * 8 + 4);

  const float* supb = support + (size_t)b * QTOT * HDIM;
  const float* qryb = query   + (size_t)b * QTOT * HDIM;

  // ---- Phase 1: LayerNorm 100 rows (one row per wave per iteration) ----
  for (int row = wave; row < 2 * QTOT; row += 8) {
    const float* src = (row < QTOT) ? (supb + row * HDIM)
                                    : (qryb + (row - QTOT) * HDIM);
    float4 a0 = *(const float4*)(src + lane * 8);
    float4 a1 = *(const float4*)(src + lane * 8 + 4);
    float s = a0.x + a0.y + a0.z + a0.w + a1.x + a1.y + a1.z + a1.w;
    float mean = wave_sum(s) * (1.0f / HDIM);
    float d[8] = {a0.x - mean, a0.y - mean, a0.z - mean, a0.w - mean,
                  a1.x - mean, a1.y - mean, a1.z - mean, a1.w - mean};
    float sq = 0.f;
#pragma unroll
    for (int j = 0; j < 8; ++j) sq += d[j] * d[j];
    float var  = wave_sum(sq) * (1.0f / HDIM);
    float rstd = rsqrtf(var + 1e-5f);
    float o[8];
    o[0] = d[0] * rstd * g0.x + e0.x;  o[1] = d[1] * rstd * g0.y + e0.y;
    o[2] = d[2] * rstd * g0.z + e0.z;  o[3] = d[3] * rstd * g0.w + e0.w;
    o[4] = d[4] * rstd * g1.x + e1.x;  o[5] = d[5] * rstd * g1.y + e1.y;
    o[6] = d[6] * rstd * g1.z + e1.z;  o[7] = d[7] * rstd * g1.w + e1.w;
    float* dst = (row < QTOT) ? &sm[OFF_SN + row * SQ]
                              : &sm[OFF_QN + (row - QTOT) * SQ];
    float nq = 0.f;
#pragma unroll
    for (int j = 0; j < 8; ++j) { dst[lane * 8 + j] = o[j]; nq += o[j] * o[j]; }
    nq = wave_sum(nq);
    if (row >= QTOT && lane == 0) sm[OFF_QQ + (row - QTOT)] = nq;  // ||q||^2
  }
  __syncthreads();

  // ---- Phase 2: mean prototypes M[n][h] and ||M||^2 (n = iter, h = tid) ----
  for (int n = 0; n < 16; ++n) {
    const int h = tid;
    float m = 0.f;
    if (n < NCLS) {
#pragma unroll
      for (int k = 0; k < KSH; ++k) m += sm[OFF_SN + (n * KSH + k) * SQ + h];
      m *= (1.0f / KSH);
      float c2 = wave_sum(m * m);
      if (lane == 0) atomicAdd(&sm[OFF_MM + n], c2);
    }
    sm[OFF_M + n * SQ + h] = m;  // rows 10..15 zero (B-operand padding)
  }
  __syncthreads();

  // ---- Phase 3: dQM = q.M^T (waves 0-3), dSM = s.M^T (waves 4-7) ----
  {
    const float* A = (wave < 4) ? &sm[OFF_QN] : &sm[OFF_SN];
    float*       D = (wave < 4) ? &sm[OFF_DQM] : &sm[OFF_DSM];
    gemm16(A, &sm[OFF_M], D, wave & 3, lane);
  }
  __syncthreads();

  // ---- Phase 4: nearest prototype per query: argmax 2*dot - qq - MM ----
  if (tid < QTOT) {
    const float qq = sm[OFF_QQ + tid];
    float best = -3.4e38f; int bi = 0;
    for (int n = 0; n < NCLS; ++n) {
      float v = 2.f * sm[OFF_DQM + tid * 16 + n] - qq - sm[OFF_MM + n];
      if (v > best) { best = v; bi = n; }
    }
    Cq[tid] = bi;
  }
  __syncthreads();

  // ---- Phase 5: counts + softmax-over-N weights per slot ----
  if (tid < NCLS) {
    int c = 0;
    for (int qi = 0; qi < QTOT; ++qi) c += (Cq[qi] == tid);
    sm[OFF_INV + tid] = 1.0f / (float)(KSH + c);
  }
  if (tid >= 64 && tid < 64 + KSH + QTOT) {
    const int s = tid - 64;                       // slot 0..54
    float raw[NCLS];
    if (s < KSH) {
      for (int n = 0; n < NCLS; ++n)
        raw[n] = sm[OFF_DSM + (n * KSH + s) * 16 + n];   // dot(s[n,s], M[n])
    } else {
      const int qi = s - KSH, c = Cq[qi];
      for (int n = 0; n < NCLS; ++n) raw[n] = 0.f;       // zero-padded rows
      raw[c] = sm[OFF_DQM + qi * 16 + c];
    }
    float mx = raw[0];
    for (int n = 1; n < NCLS; ++n) mx = fmaxf(mx, raw[n]);
    float ex[NCLS], sum = 0.f;
    for (int n = 0; n < NCLS; ++n) { ex[n] = __expf(raw[n] - mx); sum += ex[n]; }
    const float inv = 1.0f / sum;
    for (int n = 0; n < NCLS; ++n) sm[OFF_W + s * 16 + n] = ex[n] * inv;
  }
  __syncthreads();

  // ---- Phase 6: rectified prototypes P[n][h] and ||P||^2 ----
  for (int n = 0; n < 16; ++n) {
    const int h = tid;
    float acc = 0.f;
    if (n < NCLS) {
#pragma unroll
      for (int s = 0; s < KSH; ++s)
        acc += sm[OFF_W + s * 16 + n] * sm[OFF_SN + (n * KSH + s) * SQ + h];
      for (int qi = 0; qi < QTOT; ++qi) {
        float w = (Cq[qi] == n) ? sm[OFF_W + (KSH + qi) * 16 + n] : 0.f;
        acc += w * sm[OFF_QN + qi * SQ + h];
      }
      acc *= sm[OFF_INV + n];
      float p2 = wave_sum(acc * acc);
      if (lane == 0) atomicAdd(&sm[OFF_PP + n], p2);
    }
    sm[OFF_P + n * SQ + h] = acc;  // rows 10..15 zero
  }
  __syncthreads();

  // ---- Phase 7: dQP = q.P^T (waves 0-3) ----
  if (wave < 4) gemm16(&sm[OFF_QN], &sm[OFF_P], &sm[OFF_DQP], wave, lane);
  __syncthreads();

  // ---- Phase 8: logits + appended (min-1) + pred ----
  if (tid < QTOT) {
    const float qq = sm[OFF_QQ + tid];
    float mn = 3.4e38f, best = -3.4e38f; int bi = 0;
    float L[NCLS];
    for (int n = 0; n < NCLS; ++n) {
      L[n] = 2.f * sm[OFF_DQP + tid * 16 + n] - qq - sm[OFF_PP + n];
      mn = fminf(mn, L[n]);
      if (L[n] > best) { best = L[n]; bi = n; }
    }
    float* op = out + ((size_t)b * QTOT + tid) * (NCLS + 1);
    for (int n = 0; n < NCLS; ++n) op[n] = L[n];
    op[NCLS] = mn - 1.0f;
    out[(size_t)B * QTOT * (NCLS + 1) + (size_t)b * QTOT + tid] = (float)bi;
  }
}

extern "C" void kernel_launch(void* const* d_in, const int* in_sizes, int n_in,
                              void* d_out, int out_size, void* d_ws, size_t ws_size,
                              hipStream_t stream) {
  (void)n_in; (void)out_size; (void)d_ws; (void)ws_size;
  const float* support = (const float*)d_in[0];
  const float* query   = (const float*)d_in[1];
  const float* gamma   = (const float*)d_in[2];
  const float* beta    = (const float*)d_in[3];
  const int B = in_sizes[0] / (QTOT * HDIM);   // 512
  const size_t smem = (size_t)SMEM_FLOATS * sizeof(float);  // ~150 KB (<320 KB/WGP)
  PR_23965917511945_kernel<<<dim3(B), dim3(256), smem, stream>>>(
      support, query, gamma, beta, (float*)d_out, B);
}